// MetaSelectInput_61409442398235
// MI455X (gfx1250) — compile-verified
//
#include <hip/hip_runtime.h>
#include <hip/hip_bf16.h>
#include <stdint.h>

// ROI-Align (tf crop_and_resize, bilinear, 7x7) over a 5-level FPN, MI455X/gfx1250.
//
// Roofline: 200MB output (write-once) + ~170MB pyramid (gather source, L2-resident in
// the 192MB L2) => ~370MB of HBM traffic @ 23.3TB/s ~= 16us. Pure data movement; the
// op has no shared-K reduction so WMMA does not apply. The CDNA5 feature that matches
// the access pattern is the Tensor Data Mover in *gather mode*: each bilinear tap is a
// contiguous 1KB channel vector fm[b,y,x,0:256], and one TDM descriptor DMAs 16
// indexed rows (16-bit row index y*W+x, max H*W=16384) straight into LDS.
//
// Structure: 1 block per (roi,level); 7 chunks of 8 pooled positions; 2 TDM gather
// descriptors per chunk double-buffered in 2x32KB LDS; wave0 issues descriptors and
// gates on s_wait_tensorcnt while 8 wave32s blend the previous chunk from LDS and
// stream results out with non-temporal b128 stores (protecting the pyramid in L2).

typedef unsigned int u32;
typedef __attribute__((ext_vector_type(4))) u32   v4u;
typedef __attribute__((ext_vector_type(8))) int   v8i;
typedef __attribute__((ext_vector_type(4))) int   v4i;
typedef __attribute__((ext_vector_type(4))) float v4f;   // clang ext vector: valid for
                                                         // __builtin_nontemporal_store

__device__ __forceinline__ int rfl(int x) { return __builtin_amdgcn_readfirstlane(x); }

// One TDM gather: n_valid rows (each 256 f32 = 1KB) from a [H*W, 256] row view of
// fm[b], row r = y*W + x, landing densely at LDS byte offset lds_byte_off.
__device__ __forceinline__ void tdm_gather_rows(u32 lds_byte_off,
                                                const float* row_base,
                                                u32 n_rows_tensor,
                                                const u32 idx[16],
                                                u32 n_valid)
{
  uint64_t ga = (uint64_t)(uintptr_t)row_base;

  v4u g0;
  g0.x = (u32)rfl((int)(1u | (1u << 31)));                 // count=1 | gather_mode, 16-bit idx
  g0.y = (u32)rfl((int)lds_byte_off);                      // lds_addr (bytes)
  g0.z = (u32)rfl((int)(u32)(ga & 0xFFFFFFFFu));           // global_addr[31:0]
  g0.w = (u32)rfl((int)(((u32)(ga >> 32) & 0x01FFFFFFu) | 0x80000000u)); // addr[56:32]|type=2

  v8i g1;
  g1.s0 = rfl((int)(2u << 16));                            // data_size = 2 (4 bytes)
  g1.s1 = rfl((int)(256u << 16));                          // tensor_dim0 = 256 (bits 79:48 lo16)
  g1.s2 = rfl((int)((n_rows_tensor & 0xFFFFu) << 16));     // dim0 hi16=0 | tensor_dim1 lo16
  g1.s3 = rfl((int)((n_rows_tensor >> 16) | (256u << 16)));// tensor_dim1 hi16 | tile_dim0=256
  g1.s4 = rfl((int)(n_valid & 0xFFFFu));                   // tile_dim1 = #valid gather indices
  g1.s5 = rfl((int)256);                                   // tensor_dim0_stride = 256
  g1.s6 = 0;
  g1.s7 = 0;

  v4i g2, g3;                                              // packed 16-bit row indices
  g2.x = rfl((int)(idx[0]  | (idx[1]  << 16)));
  g2.y = rfl((int)(idx[2]  | (idx[3]  << 16)));
  g2.z = rfl((int)(idx[4]  | (idx[5]  << 16)));
  g2.w = rfl((int)(idx[6]  | (idx[7]  << 16)));
  g3.x = rfl((int)(idx[8]  | (idx[9]  << 16)));
  g3.y = rfl((int)(idx[10] | (idx[11] << 16)));
  g3.z = rfl((int)(idx[12] | (idx[13] << 16)));
  g3.w = rfl((int)(idx[14] | (idx[15] << 16)));

#if defined(__clang_major__) && (__clang_major__ >= 23)
  v8i zero8 = {0, 0, 0, 0, 0, 0, 0, 0};
  __builtin_amdgcn_tensor_load_to_lds(g0, g1, g2, g3, zero8, 0);
#else
  __builtin_amdgcn_tensor_load_to_lds(g0, g1, g2, g3, 0);
#endif
}

#define POOL      7
#define NPOS      49
#define NCHUNK    7            // 7 chunks x 8 positions (last chunk: 1 position)
#define BUF_BYTES 32768u       // 8 positions x 4 taps x 1KB
#define ROI_ELEMS 50176000ull  // 800*49*1280

__global__ void __launch_bounds__(256)
roi_crop_resize_tdm(const float* __restrict__ boxes,   // [800,4] (y1,x1,y2,x2) in pixels
                    const float* __restrict__ fm0,
                    const float* __restrict__ fm1,
                    const float* __restrict__ fm2,
                    const float* __restrict__ fm3,
                    const float* __restrict__ fm4,
                    float* __restrict__ out)           // rois flat ++ box_ids flat
{
  extern __shared__ float lds[];                       // 2 x 32KB double buffer

  const int blk = blockIdx.x;                          // 800*5 blocks
  const int n   = blk / 5;                             // roi 0..799
  const int lvl = blk - n * 5;                         // pyramid level 0..4
  const int b   = n / 100;                             // batch index == box_id

  int S; const float* fm;
  switch (lvl) {
    case 0:  S = 8;   fm = fm0; break;
    case 1:  S = 16;  fm = fm1; break;
    case 2:  S = 32;  fm = fm2; break;
    case 3:  S = 64;  fm = fm3; break;
    default: S = 128; fm = fm4; break;
  }
  const int   H   = 1024 / S;                          // H == W
  const float Hm1 = (float)(H - 1);
  const float* fmb = fm + (size_t)b * (size_t)(H * H) * 256u;

  const float by1 = boxes[4 * n + 0], bx1 = boxes[4 * n + 1];
  const float by2 = boxes[4 * n + 2], bx2 = boxes[4 * n + 3];
  const float inv = 1.0f / ((float)S * Hm1);           // == normalize then *(H-1)
  const float ny1 = by1 * inv, nx1 = bx1 * inv;
  const float ny2 = by2 * inv, nx2 = bx2 * inv;

  const int wave = threadIdx.x >> 5;
  const int lane = threadIdx.x & 31;

  // Build + issue one 16-row (4-position) gather descriptor: chunk c, half h.
  auto issue_desc = [&](int c, int half, u32 n_valid) {
    u32 idx[16];
#pragma unroll
    for (int k = 0; k < 16; ++k) {
      int p = c * 8 + half * 4 + (k >> 2); if (p > NPOS - 1) p = NPOS - 1;
      const int i = p / POOL, j = p - POOL * i;
      const float ys = (ny1 + (ny2 - ny1) * ((float)i * (1.0f / 6.0f))) * Hm1;
      const float xs = (nx1 + (nx2 - nx1) * ((float)j * (1.0f / 6.0f))) * Hm1;
      const int y0f = (int)floorf(ys), x0f = (int)floorf(xs);
      const int y0 = min(max(y0f,     0), H - 1);
      const int y1 = min(max(y0f + 1, 0), H - 1);
      const int x0 = min(max(x0f,     0), H - 1);
      const int x1 = min(max(x0f + 1, 0), H - 1);
      const int yy = (k & 2) ? y1 : y0;
      const int xx = (k & 1) ? x1 : x0;
      idx[k] = (u32)(yy * H + xx);
    }
    tdm_gather_rows((u32)(c & 1) * BUF_BYTES + (u32)half * (BUF_BYTES / 2u),
                    fmb, (u32)(H * H), idx, n_valid);
  };

  // ---- software pipeline: TDM fills buf[(c+1)&1] while waves blend buf[c&1] ----
  if (threadIdx.x < 32) { issue_desc(0, 0, 16u); issue_desc(0, 1, 16u); }

  const v4f* lv = (const v4f*)lds;

  for (int c = 0; c < NCHUNK; ++c) {
    if (threadIdx.x < 32) {
      if (c + 1 < NCHUNK - 1) {          // next chunk is full: 2 descriptors
        issue_desc(c + 1, 0, 16u);
        issue_desc(c + 1, 1, 16u);
        __builtin_amdgcn_s_wait_tensorcnt((short)2);  // current chunk's 2 descs done
      } else if (c + 1 == NCHUNK - 1) {  // tail chunk: position 48 only -> 1 descriptor
        issue_desc(c + 1, 0, 4u);
        __builtin_amdgcn_s_wait_tensorcnt((short)1);
      } else {
        __builtin_amdgcn_s_wait_tensorcnt((short)0);
      }
    }
    __syncthreads();                                   // chunk c tap rows visible in LDS

    // 8 wave32s, one pooled position each; lane covers channels 4q..4q+3, q in {lane, lane+32}.
    const int p = c * 8 + wave;
    if (p < NPOS) {
      const int i = p / POOL, j = p - POOL * i;
      const float ys = (ny1 + (ny2 - ny1) * ((float)i * (1.0f / 6.0f))) * Hm1;
      const float xs = (nx1 + (nx2 - nx1) * ((float)j * (1.0f / 6.0f))) * Hm1;
      const float wy = ys - floorf(ys);
      const float wx = xs - floorf(xs);
      const bool valid = (ys >= 0.0f) && (ys <= Hm1) && (xs >= 0.0f) && (xs <= Hm1);

      // position's 4 tap rows at v4f offset: buf*(2048) + wave*256 (+ tap*64)
      const int rbase = (c & 1) * 2048 + wave * 256;
      float* op = out + ((size_t)n * NPOS + (size_t)p) * 1280 + (size_t)lvl * 256;

#pragma unroll
      for (int h = 0; h < 2; ++h) {
        const int q = lane + 32 * h;
        v4f a  = lv[rbase +   0 + q];   // (y0,x0)
        v4f bb = lv[rbase +  64 + q];   // (y0,x1)
        v4f cc = lv[rbase + 128 + q];   // (y1,x0)
        v4f dd = lv[rbase + 192 + q];   // (y1,x1)

        v4f r; float top, bot;
        top = a.x + wx * (bb.x - a.x); bot = cc.x + wx * (dd.x - cc.x);
        r.x = valid ? top + wy * (bot - top) : 0.0f;
        top = a.y + wx * (bb.y - a.y); bot = cc.y + wx * (dd.y - cc.y);
        r.y = valid ? top + wy * (bot - top) : 0.0f;
        top = a.z + wx * (bb.z - a.z); bot = cc.z + wx * (dd.z - cc.z);
        r.z = valid ? top + wy * (bot - top) : 0.0f;
        top = a.w + wx * (bb.w - a.w); bot = cc.w + wx * (dd.w - cc.w);
        r.w = valid ? top + wy * (bot - top) : 0.0f;

        // Write-once streaming output: non-temporal so the FPN stays resident in L2.
        __builtin_nontemporal_store(r, (v4f*)(op + 4 * q));
      }
    }
    __syncthreads();                                   // buf[c&1] free before refill at c+2
  }

  // tuple output #2: box_ids = repeat(arange(8), 100), appended after the rois
  if (lvl == 0 && threadIdx.x == 0) {
    out[ROI_ELEMS + (size_t)n] = (float)b;
  }
}

extern "C" void kernel_launch(void* const* d_in, const int* in_sizes, int n_in,
                              void* d_out, int out_size, void* d_ws, size_t ws_size,
                              hipStream_t stream) {
  const float* boxes = (const float*)d_in[0];   // [8,100,4]
  const float* fm0   = (const float*)d_in[1];   // [8,128,128,256]
  const float* fm1   = (const float*)d_in[2];   // [8, 64, 64,256]
  const float* fm2   = (const float*)d_in[3];   // [8, 32, 32,256]
  const float* fm3   = (const float*)d_in[4];   // [8, 16, 16,256]
  const float* fm4   = (const float*)d_in[5];   // [8,  8,  8,256]
  float* out = (float*)d_out;

  const dim3 grid(800 * 5);
  const dim3 block(256);
  const size_t shmem = 2u * BUF_BYTES;          // 64KB: double-buffered gather tiles

  roi_crop_resize_tdm<<<grid, block, shmem, stream>>>(boxes, fm0, fm1, fm2, fm3, fm4, out);
}